// SphericalSpectralTimeConv_34677565948283
// MI455X (gfx1250) — compile-verified
//
#include <hip/hip_runtime.h>
#include <hip/hip_bf16.h>

typedef __attribute__((ext_vector_type(2))) float v2f;
typedef __attribute__((ext_vector_type(8))) float v8f;

#define PI_D     3.14159265358979323846
#define TWO_PI_D 6.28318530717958647692

#define LI  256   // L_IN == L_OUT
#define MI  511   // 2*L_IN - 1
#define LF  128   // L_FREQ
#define MF  255   // 2*L_FREQ - 1
#define NC  32    // channels
#define NC2 64    // re|im packed channels

// ---------------------------------------------------------------------------
// Legendre basis P[mm][l][t]  (mm = |m|, 256 x 256 x 256), fp32, double recur.
// Pfull[l, 255+m, t] = P[mm][l][t];  Pfull[l, 255-m, t] = (-1)^m P  (m>0)
// ---------------------------------------------------------------------------
__global__ void gen_P(float* __restrict__ P) {
  int t = threadIdx.x;        // theta index 0..255
  int mm = blockIdx.x;        // |m| 0..255
  double theta = PI_D * (2.0 * t + 1.0) / (double)MI;
  double ct = cos(theta), st = sin(theta);
  double pmm = sqrt(1.0 / (4.0 * PI_D));
  for (int m = 1; m <= mm; ++m)
    pmm = -sqrt((2.0 * m + 1.0) / (2.0 * m)) * st * pmm;
  float* Pm = P + (size_t)mm * (LI * LI);
  for (int l = 0; l < mm; ++l) Pm[l * LI + t] = 0.0f;
  Pm[mm * LI + t] = (float)pmm;
  double pl2 = pmm, pl1 = pmm;
  if (mm + 1 < LI) {
    pl1 = sqrt(2.0 * mm + 3.0) * ct * pmm;
    Pm[(mm + 1) * LI + t] = (float)pl1;
  }
  for (int l = mm + 2; l < LI; ++l) {
    double ll = (double)l * l, mm2 = (double)mm * mm;
    double a = sqrt((4.0 * ll - 1.0) / (ll - mm2));
    double b = sqrt((2.0 * l + 1.0) * (l - 1.0 - mm) * (l - 1.0 + mm) /
                    ((2.0 * l - 3.0) * (ll - mm2)));
    double pl = a * ct * pl1 - b * pl2;
    Pm[l * LI + t] = (float)pl;
    pl2 = pl1; pl1 = pl;
  }
}

// Forward DFT matrices (m-shifted):  Ecos/Ensin [511][512]
// row j (0..510) <-> FFT bin k=(j+256)%511; exp(-i 2pi k p /511)
__global__ void gen_fwd_dft(float* __restrict__ Ecos, float* __restrict__ Ensin) {
  int p = blockIdx.x * blockDim.x + threadIdx.x;  // 0..511
  int j = blockIdx.y;                              // 0..510
  if (p >= 512) return;
  size_t idx = (size_t)j * 512 + p;
  if (p == MI) { Ecos[idx] = 0.0f; Ensin[idx] = 0.0f; return; }
  int k = (j + 256) % MI;
  long r = ((long)k * p) % MI;
  double th = TWO_PI_D * (double)r / (double)MI;
  Ecos[idx]  = (float)cos(th);
  Ensin[idx] = (float)(-sin(th));
}

// Inverse DFT matrices: Icos/Insin [511][256]; col j2 (0..254) <-> m=j2-127
__global__ void gen_inv_dft(float* __restrict__ Icos, float* __restrict__ Insin) {
  int j2 = threadIdx.x;  // 0..255
  int p  = blockIdx.x;   // 0..510
  size_t idx = (size_t)p * 256 + j2;
  if (j2 >= MF) { Icos[idx] = 0.0f; Insin[idx] = 0.0f; return; }
  int m = j2 - (LF - 1);
  int k = (m % MI + MI) % MI;
  long r = ((long)k * p) % MI;
  double th = TWO_PI_D * (double)r / (double)MI;
  Icos[idx]  = (float)cos(th);
  Insin[idx] = (float)(-sin(th));
}

// t_cplx: tc[0..127]=real, tc[128..255]=imag
__global__ void time_embed(const float* __restrict__ t_emb,
                           const float* __restrict__ w_tr, const float* __restrict__ b_tr,
                           const float* __restrict__ w_ti, const float* __restrict__ b_ti,
                           float* __restrict__ tc) {
  int i = threadIdx.x;            // 0..255
  int l = i & (LF - 1);
  bool im = i >= LF;
  const float* w = im ? w_ti : w_tr;
  float s = im ? b_ti[l] : b_tr[l];
  for (int k = 0; k < 256; ++k) s += t_emb[k] * w[k * LF + l];
  tc[i] = s;
}

// ---------------------------------------------------------------------------
// Generic batched GEMM, fp32 WMMA 16x16x4.  C[b](MxN) = alpha*A[b](MxK)*B[b]
// block = 128 threads (4 wave32); 64(M) x N tile; K chunked by 32 through LDS.
// Branchless clamped global->reg staging with cross-chunk prefetch; B stored
// K-pair interleaved so each fragment is one ds_load_b64; fragment loads are
// software-pipelined across the unrolled K-steps.
// AMODE: 0 alpha=1 | 1 alpha=w(theta_b) | 2 A+=|b-255|*64K, Legendre sign
//        3 A+=|b-127|*64K, Legendre sign ; TRANSA: A[row][k]=mem[k*lda+row]
// ---------------------------------------------------------------------------
#define GTK 32
template <int N, int TRANSA, int ACCUM, int AMODE>
__global__ __launch_bounds__(128) void gemm_wmma(
    const float* __restrict__ A, int lda,
    const float* __restrict__ B, int ldb, long long sB,
    float* __restrict__ C, int ldc, long long sC,
    int M, int K) {
  constexpr int NT  = N / 16;            // 2 or 4 accumulators per wave
  constexpr int NSH = (N == 64) ? 6 : 5; // log2(N)
  constexpr int RS  = 2 * N + 32;        // lB row stride (words), == 32 mod 64
  constexpr int AP  = 8;                 // A pairs per thread (64*16/128)
  constexpr int BP  = (GTK / 2) * N / 128; // B pairs per thread (4 or 8)
  __shared__ float lA[64][GTK + 4];      // stride 36: even, conflict-free
  __shared__ float lBf[(GTK / 2) * RS];
  int b  = blockIdx.y;
  int m0 = blockIdx.x * 64;
  float alpha = 1.0f;
  const float* Ab = A;
  if (AMODE == 1) {
    float dang = (float)(TWO_PI_D / (double)MI);
    float th = (float)(PI_D * (2.0 * b + 1.0) / (double)MI);
    alpha = sinf(th) * dang * dang;                 // quadrature weight w(theta)
  } else if (AMODE == 2) {
    int m = b - (LI - 1); int mm = m < 0 ? -m : m;
    Ab = A + (size_t)mm * (LI * LI);
    if (m < 0 && (mm & 1)) alpha = -1.0f;           // Pfull(-m) = (-1)^m P(m)
  } else if (AMODE == 3) {
    int m = b - (LF - 1); int mm = m < 0 ? -m : m;
    Ab = A + (size_t)mm * (LI * LI);
    if (m < 0 && (mm & 1)) alpha = -1.0f;
  }
  const float* Bb = B + (size_t)b * sB;
  float* Cb = C + (size_t)b * sC;
  int tid = threadIdx.x;
  int w = tid >> 5, lane = tid & 31;
  int lhalf = (lane >> 4) << 1;   // 0 or 2 : K sub-pair selector (ISA A/B layout)
  int l15 = lane & 15;
  int mrow = (w << 4) + l15;
  int kpb = lhalf >> 1;           // 0 or 1 : K-pair index offset per half-wave

  v2f ra[AP];                     // staged A K-pairs (rows clamped; edge rows
                                  // feed tiles whose rows are never stored)
  v2f rb[BP];                     // staged B K-pairs (K-edge rows zeroed)

  auto gload = [&](int k0) {
    if (!TRANSA) {
#pragma unroll
      for (int i = 0; i < AP; ++i) {
        int e = tid + i * 128;
        int r = e >> 4, kp = e & 15;
        int gr = m0 + r;
        int gr2 = gr < M ? gr : M - 1;               // clamp: always in-bounds
        ra[i] = *(const v2f*)(Ab + (size_t)gr2 * lda + (k0 + (kp << 1)));
      }
    } else {  // A[row][k] = mem[k*lda + row]; pair of K-adjacent columns
#pragma unroll
      for (int i = 0; i < AP; ++i) {
        int e = tid + i * 128;
        int r = e & 63, kp = e >> 6;
        int gr = m0 + r;
        int gr2 = gr < M ? gr : M - 1;
        int gk = k0 + (kp << 1);
        v2f v;
        v.x = Ab[(size_t)gk * lda + gr2];
        v.y = Ab[(size_t)(gk + 1) * lda + gr2];
        ra[i] = v;
      }
    }
#pragma unroll
    for (int i = 0; i < BP; ++i) {
      int e = tid + i * 128;
      int kp = e >> NSH, n = e & (N - 1);
      int gk0 = k0 + (kp << 1), gk1 = gk0 + 1;
      int c0 = gk0 < K ? gk0 : K - 1;
      int c1 = gk1 < K ? gk1 : K - 1;
      v2f v;
      v.x = Bb[(size_t)c0 * ldb + n];
      v.y = Bb[(size_t)c1 * ldb + n];
      if (gk0 >= K) v.x = 0.0f;                      // K-edge must be zero
      if (gk1 >= K) v.y = 0.0f;
      rb[i] = v;
    }
  };

  auto lstore = [&]() {
    if (!TRANSA) {
#pragma unroll
      for (int i = 0; i < AP; ++i) {
        int e = tid + i * 128;
        int r = e >> 4, kp = e & 15;
        *(v2f*)&lA[r][kp << 1] = alpha * ra[i];
      }
    } else {
#pragma unroll
      for (int i = 0; i < AP; ++i) {
        int e = tid + i * 128;
        int r = e & 63, kp = e >> 6;
        *(v2f*)&lA[r][kp << 1] = alpha * ra[i];
      }
    }
#pragma unroll
    for (int i = 0; i < BP; ++i) {
      int e = tid + i * 128;
      int kp = e >> NSH, n = e & (N - 1);
      *(v2f*)&lBf[kp * RS + (n << 1)] = rb[i];       // K-pair interleaved
    }
  };

  v8f acc[NT] = {};
  gload(0);
  for (int k0 = 0; k0 < K; k0 += GTK) {
    lstore();
    __syncthreads();
    if (k0 + GTK < K) gload(k0 + GTK);  // prefetch next chunk under compute
    v2f af[2];
    v2f bf[2][NT];
    auto ldfrag = [&](int kk, int s) {
      af[s] = *(const v2f*)&lA[mrow][kk + lhalf];
      int kp = (kk >> 1) + kpb;
#pragma unroll
      for (int nt = 0; nt < NT; ++nt)
        bf[s][nt] = *(const v2f*)&lBf[kp * RS + (((nt << 4) + l15) << 1)];
    };
    ldfrag(0, 0);
#pragma unroll
    for (int kk = 0; kk < GTK; kk += 4) {
      int cur = (kk >> 2) & 1;
      if (kk + 4 < GTK) ldfrag(kk + 4, cur ^ 1);
#pragma unroll
      for (int nt = 0; nt < NT; ++nt)
        acc[nt] = __builtin_amdgcn_wmma_f32_16x16x4_f32(
            false, af[cur], false, bf[cur][nt], (short)0, acc[nt], false, false);
    }
    __syncthreads();
  }
#pragma unroll
  for (int nt = 0; nt < NT; ++nt) {
    int col = (nt << 4) + l15;
#pragma unroll
    for (int r = 0; r < 8; ++r) {
      int grow = m0 + (w << 4) + r + ((lane >> 4) << 3);
      if (grow < M) {
        size_t idx = (size_t)grow * ldc + col;
        float v = acc[nt][r];
        if (ACCUM) v += Cb[idx];
        Cb[idx] = v;
      }
    }
  }
}

// ---------------------------------------------------------------------------
// jax.image.resize bilinear (antialias, triangle/kernel_scale, per-dim norm)
// flm[j 511][l 256][c2 64]  ->  flmR[l2 128][j2 255][c2 64]
// ---------------------------------------------------------------------------
__global__ void resize_kernel(const float* __restrict__ flm, float* __restrict__ flmR) {
  int idx = blockIdx.x * blockDim.x + threadIdx.x;
  if (idx >= LF * MF * NC2) return;
  int c  = idx & 63;
  int j2 = (idx >> 6) % MF;
  int l2 = idx / (MF * NC2);
  const float ksl = (float)LI / (float)LF;   // 2.0
  const float ksm = (float)MI / (float)MF;   // 511/255
  float sl = (l2 + 0.5f) * ksl - 0.5f;
  float sm = (j2 + 0.5f) * ksm - 0.5f;
  float wl[8]; int il[8]; int nl = 0; float swl = 0.0f;
  int lo = (int)ceilf(sl - ksl), hi = (int)floorf(sl + ksl);
  for (int i = lo; i <= hi; ++i) {
    if (i < 0 || i >= LI) continue;
    float wg = 1.0f - fabsf((float)i - sl) / ksl;
    if (wg <= 0.0f || nl >= 8) continue;
    wl[nl] = wg; il[nl] = i; swl += wg; ++nl;
  }
  float wm[8]; int im[8]; int nm = 0; float swm = 0.0f;
  lo = (int)ceilf(sm - ksm); hi = (int)floorf(sm + ksm);
  for (int i = lo; i <= hi; ++i) {
    if (i < 0 || i >= MI) continue;
    float wg = 1.0f - fabsf((float)i - sm) / ksm;
    if (wg <= 0.0f || nm >= 8) continue;
    wm[nm] = wg; im[nm] = i; swm += wg; ++nm;
  }
  float out = 0.0f;
  for (int a = 0; a < nl; ++a)
    for (int q = 0; q < nm; ++q)
      out += wl[a] * wm[q] * flm[(size_t)im[q] * (LI * NC2) + (size_t)il[a] * NC2 + c];
  flmR[(size_t)l2 * (MF * NC2) + (size_t)j2 * NC2 + c] = out / (swl * swm);
}

// complex channel mix: flmW[l][m][o|32+o] = flmR[l][m][i|32+i] x (tc[l]*(wr+i wi))
__global__ void spectral_conv(const float* __restrict__ flmR,
                              const float* __restrict__ wr, const float* __restrict__ wi,
                              const float* __restrict__ tc, float* __restrict__ flmW) {
  int idx = blockIdx.x * blockDim.x + threadIdx.x;
  if (idx >= LF * MF * NC) return;
  int o = idx & 31;
  int m = (idx >> 5) % MF;
  int l = idx / (MF * NC);
  float tr = tc[l], ti = tc[LF + l];
  const float* f  = flmR + (size_t)l * (MF * NC2) + (size_t)m * NC2;
  const float* Wr = wr + (size_t)l * (NC * NC) + o;
  const float* Wi = wi + (size_t)l * (NC * NC) + o;
  float orr = 0.0f, oii = 0.0f;
  for (int i = 0; i < NC; ++i) {
    float fre = f[i], fim = f[NC + i];
    float wre = Wr[i * NC], wim = Wi[i * NC];
    float Wre = tr * wre - ti * wim;
    float Wim = tr * wim + ti * wre;
    orr += fre * Wre - fim * Wim;
    oii += fre * Wim + fim * Wre;
  }
  float* out = flmW + (size_t)l * (MF * NC2) + (size_t)m * NC2;
  out[o] = orr;
  out[NC + o] = oii;
}

extern "C" void kernel_launch(void* const* d_in, const int* in_sizes, int n_in,
                              void* d_out, int out_size, void* d_ws, size_t ws_size,
                              hipStream_t stream) {
  const float* x     = (const float*)d_in[0];  // (256,511,32)
  const float* t_emb = (const float*)d_in[1];  // (256,)
  const float* wreal = (const float*)d_in[2];  // (128,32,32)
  const float* wimag = (const float*)d_in[3];
  const float* w_tr  = (const float*)d_in[4];  // (256,128)
  const float* b_tr  = (const float*)d_in[5];
  const float* w_ti  = (const float*)d_in[6];
  const float* b_ti  = (const float*)d_in[7];
  float* out = (float*)d_out;                  // (256,511,32)

  float* ws = (float*)d_ws;
  size_t off = 0;
  float* P     = ws + off; off += (size_t)256 * 256 * 256;  // 16.78M
  float* Ecos  = ws + off; off += (size_t)MI * 512;
  float* Ensin = ws + off; off += (size_t)MI * 512;
  float* Icos  = ws + off; off += (size_t)MI * 256;
  float* Insin = ws + off; off += (size_t)MI * 256;
  float* F     = ws + off; off += (size_t)MI * LI * NC2;    // wF [j][t][c2]
  float* flm   = ws + off; off += (size_t)MI * LI * NC2;    // [j][l][c2]
  float* flmR  = ws + off; off += (size_t)LF * MF * NC2;    // [l][m][c2]
  float* flmW  = ws + off; off += (size_t)LF * MF * NC2;
  float* G     = ws + off; off += (size_t)LI * MF * NC2;    // [t][j2][c2]
  float* tc    = ws + off; off += 256;
  (void)ws_size; (void)in_sizes; (void)n_in; (void)out_size;

  // basis + DFT matrix generation (all deterministic, regenerated each call)
  gen_P<<<dim3(256), dim3(256), 0, stream>>>(P);
  gen_fwd_dft<<<dim3(2, MI), dim3(256), 0, stream>>>(Ecos, Ensin);
  gen_inv_dft<<<dim3(MI), dim3(256), 0, stream>>>(Icos, Insin);
  time_embed<<<dim3(1), dim3(256), 0, stream>>>(t_emb, w_tr, b_tr, w_ti, b_ti, tc);

  // forward DFT (quadrature weight w(theta_t) folded in via AMODE=1)
  // F[j][t][0..31] = w_t * sum_p cos * x ;  F[j][t][32..63] = w_t * sum_p -sin * x
  gemm_wmma<32, 0, 0, 1><<<dim3(8, LI), dim3(128), 0, stream>>>(
      Ecos, 512, x, NC, (long long)MI * NC, F, LI * NC2, 64LL, MI, MI);
  gemm_wmma<32, 0, 0, 1><<<dim3(8, LI), dim3(128), 0, stream>>>(
      Ensin, 512, x, NC, (long long)MI * NC, F + NC, LI * NC2, 64LL, MI, MI);

  // forward Legendre: flm[j][l][c2] = sign_j * P[|j-255|] (l,t) x F[j](t,c2)
  gemm_wmma<64, 0, 0, 2><<<dim3(4, MI), dim3(128), 0, stream>>>(
      P, LI, F, NC2, (long long)LI * NC2, flm, NC2, (long long)LI * NC2, LI, LI);

  // bilinear resize (jax antialias) + time-modulated complex channel mix
  resize_kernel<<<dim3((LF * MF * NC2 + 255) / 256), dim3(256), 0, stream>>>(flm, flmR);
  spectral_conv<<<dim3((LF * MF * NC + 255) / 256), dim3(256), 0, stream>>>(
      flmR, wreal, wimag, tc, flmW);

  // inverse Legendre: G[t][j2][c2] = sign * P[|j2-127|]^T (t,l) x flmW[j2](l,c2)
  gemm_wmma<64, 1, 0, 3><<<dim3(4, MF), dim3(128), 0, stream>>>(
      P, LI, flmW, MF * NC2, 64LL, G, MF * NC2, 64LL, LI, LF);

  // inverse DFT (real part):  out[t][p][c] = Icos x Gr  +  Insin x Gi
  gemm_wmma<32, 0, 0, 0><<<dim3(8, LI), dim3(128), 0, stream>>>(
      Icos, 256, G, NC2, (long long)MF * NC2, out, NC, (long long)MI * NC, MI, MF);
  gemm_wmma<32, 0, 1, 0><<<dim3(8, LI), dim3(128), 0, stream>>>(
      Insin, 256, G + NC, NC2, (long long)MF * NC2, out, NC, (long long)MI * NC, MI, MF);
}